// GATActorCritic_39204461478860
// MI455X (gfx1250) — compile-verified
//
#include <hip/hip_runtime.h>
#include <math.h>

// ---------------- problem constants ----------------
#define NNODES 100000
#define NEDGES 800000
#define INDIM  7
#define HID    64
#define HEADS  4
#define NCLUST 4
#define NEG_SLOPE 0.2f
#define MTILES ((NNODES + 15) / 16)   // 6250 exactly

typedef float v2f __attribute__((ext_vector_type(2)));
typedef float v8f __attribute__((ext_vector_type(8)));

// ---- float atomic-max via order-preserving uint mapping ----
__device__ __forceinline__ unsigned f2ord(float f) {
    unsigned u = __float_as_uint(f);
    return (u & 0x80000000u) ? ~u : (u | 0x80000000u);
}
__device__ __forceinline__ float ord2f(unsigned u) {
    return (u & 0x80000000u) ? __uint_as_float(u & 0x7fffffffu)
                             : __uint_as_float(~u);
}
__device__ __forceinline__ float leaky(float v) {
    return v > 0.f ? v : NEG_SLOPE * v;
}

// ---------------- K0: edge-scalar precompute ----------------
// a_e[e][h] = edge_attr[e] * ce[h],  ce[h] = dot(We_head_h, att_e_head_h)
__global__ void k_setup(const float* __restrict__ We1, const float* __restrict__ ae1,
                        const float* __restrict__ We2, const float* __restrict__ ae2,
                        float* __restrict__ ce1, float* __restrict__ ce2) {
    int t = threadIdx.x;
    if (t < HEADS) {
        float s = 0.f;
        for (int d = 0; d < HID; ++d) s += We1[t * HID + d] * ae1[t * HID + d];
        ce1[t] = s;
    } else if (t == HEADS) {
        float s = 0.f;
        for (int d = 0; d < HID; ++d) s += We2[d] * ae2[d];
        ce2[0] = s;
    }
}

// ---------------- K1: conv1 node phase ----------------
// wave per node: xh1 = x @ W1 (K=7), a_src1/a_dst1 per head, zero conv1 accumulators
__global__ void k_node1(const float* __restrict__ x,  const float* __restrict__ W1,
                        const float* __restrict__ as1, const float* __restrict__ ad1,
                        float* __restrict__ xh1,  float* __restrict__ acc1,
                        float* __restrict__ asrc1, float* __restrict__ adst1,
                        float* __restrict__ dsum1, unsigned* __restrict__ dmax1) {
    int n    = (blockIdx.x * blockDim.x + threadIdx.x) >> 5;
    int lane = threadIdx.x & 31;
    if (n >= NNODES) return;

    float xv[INDIM];
#pragma unroll
    for (int k = 0; k < INDIM; ++k) xv[k] = x[(size_t)n * INDIM + k];

    float ps[4] = {0.f, 0.f, 0.f, 0.f};
    float pd[4] = {0.f, 0.f, 0.f, 0.f};
    size_t base = (size_t)n * 256;
#pragma unroll
    for (int j = 0; j < 8; ++j) {
        int c = lane + 32 * j;            // head = c>>6 = j>>1
        float s = 0.f;
#pragma unroll
        for (int k = 0; k < INDIM; ++k) s += xv[k] * W1[k * 256 + c];
        xh1[base + c]  = s;
        acc1[base + c] = 0.f;             // zero message accumulator
        ps[j >> 1] += s * as1[c];
        pd[j >> 1] += s * ad1[c];
    }
#pragma unroll
    for (int off = 16; off; off >>= 1) {
#pragma unroll
        for (int h = 0; h < 4; ++h) {
            ps[h] += __shfl_xor(ps[h], off, 32);
            pd[h] += __shfl_xor(pd[h], off, 32);
        }
    }
    if (lane == 0) {
#pragma unroll
        for (int h = 0; h < 4; ++h) {
            asrc1[n * 4 + h] = ps[h];
            adst1[n * 4 + h] = pd[h];
            dsum1[n * 4 + h] = 0.f;
            dmax1[n * 4 + h] = 0u;        // ord(-inf) region
        }
    }
}

// ---------------- K2: conv1 edge pass A (per-dst max) ----------------
__global__ void k_edge_max1(const int* __restrict__ ei, const float* __restrict__ ea,
                            const float* __restrict__ asrc1, const float* __restrict__ adst1,
                            const float* __restrict__ ce1, unsigned* __restrict__ dmax1) {
    int tid = blockIdx.x * blockDim.x + threadIdx.x;
    if (tid >= NEDGES * 4) return;
    int e = tid >> 2, h = tid & 3;
    int s = ei[e], d = ei[NEDGES + e];
    float lg = leaky(asrc1[s * 4 + h] + adst1[d * 4 + h] + ea[e] * ce1[h]);
    atomicMax(&dmax1[d * 4 + h], f2ord(lg));
}

// ---------------- K3: conv1 edge pass B (fused exp + sum + scatter) ----------------
// wave per edge; scatters unnormalized exp * xh1[src] (normalization is linear -> done per node later)
__global__ void k_edge_msg1(const int* __restrict__ ei, const float* __restrict__ ea,
                            const float* __restrict__ asrc1, const float* __restrict__ adst1,
                            const float* __restrict__ ce1, const unsigned* __restrict__ dmax1,
                            const float* __restrict__ xh1,
                            float* __restrict__ dsum1, float* __restrict__ acc1) {
    int e    = (blockIdx.x * blockDim.x + threadIdx.x) >> 5;
    int lane = threadIdx.x & 31;
    if (e >= NEDGES) return;
    int s = ei[e], d = ei[NEDGES + e];
    float eav = ea[e];
    float w[4];
#pragma unroll
    for (int h = 0; h < 4; ++h) {
        float lg = leaky(asrc1[s * 4 + h] + adst1[d * 4 + h] + eav * ce1[h]);
        w[h] = __expf(lg - ord2f(dmax1[d * 4 + h]));
    }
    if (lane == 0) {
#pragma unroll
        for (int h = 0; h < 4; ++h) atomicAdd(&dsum1[d * 4 + h], w[h]);
    }
    size_t sb = (size_t)s * 256, db = (size_t)d * 256;
#pragma unroll
    for (int j = 0; j < 8; ++j) {
        int c = lane + 32 * j;
        atomicAdd(&acc1[db + c], xh1[sb + c] * w[j >> 1]);   // L2-resident atomics
    }
}

// ---------------- K4: conv1 normalize + bias + ELU; init conv2 state ----------------
__global__ void k_finish1(const float* __restrict__ dsum1, const float* __restrict__ b1,
                          float* __restrict__ acc1, float* __restrict__ out2,
                          float* __restrict__ dsum2, unsigned* __restrict__ dmax2,
                          float* __restrict__ zsum, float* __restrict__ cfsum,
                          float* __restrict__ cnt) {
    if (blockIdx.x == 0) {                 // zero cluster-pool accumulators once
        int t = threadIdx.x;
        zsum[t] = 0.f;                     // 256 entries, blockDim==256
        if (t < NCLUST) { cfsum[t] = 0.f; cnt[t] = 0.f; }
    }
    int n    = (blockIdx.x * blockDim.x + threadIdx.x) >> 5;
    int lane = threadIdx.x & 31;
    if (n >= NNODES) return;
    size_t base = (size_t)n * 256;
#pragma unroll
    for (int j = 0; j < 8; ++j) {
        int c = lane + 32 * j;
        float v = acc1[base + c] / dsum1[n * 4 + (j >> 1)] + b1[c];
        acc1[base + c] = v > 0.f ? v : (__expf(v) - 1.f);    // ELU, h1 in place
    }
    out2[(size_t)n * 64 + lane]      = 0.f;
    out2[(size_t)n * 64 + lane + 32] = 0.f;
    if (lane == 0) { dsum2[n] = 0.f; dmax2[n] = 0u; }
}

// ---------------- K5: conv2 GEMM  xh2[N,64] = h1[N,256] @ W2[256,64]  (WMMA f32) ----------------
// One wave owns a 16x64 output tile: 4 accumulators, 64 K-steps of V_WMMA_F32_16X16X4_F32.
// W2 staged in LDS in two 32KB phases (128 K-rows each).
__global__ void k_gemm2(const float* __restrict__ h1, const float* __restrict__ W2,
                        float* __restrict__ xh2) {
    __shared__ float sW[128 * 64];
    int wave = threadIdx.x >> 5;
    int lane = threadIdx.x & 31;
    int tile = blockIdx.x * 8 + wave;
    bool valid = tile < MTILES;
    int m0    = tile * 16;
    int mrow  = (valid ? m0 : 0) + (lane & 15);
    int khalf = (lane >> 4) * 2;          // A/B VGPR layout: lanes 16-31 hold K+2,K+3

    v8f zero8 = {0.f, 0.f, 0.f, 0.f, 0.f, 0.f, 0.f, 0.f};
    v8f acc[4] = {zero8, zero8, zero8, zero8};
    const float* hrow = h1 + (size_t)mrow * 256;

    for (int phase = 0; phase < 2; ++phase) {
        for (int i = threadIdx.x; i < 128 * 64; i += blockDim.x)
            sW[i] = W2[phase * 128 * 64 + i];
        __syncthreads();
        if (valid) {
            for (int k0 = 0; k0 < 128; k0 += 4) {
                int kk = phase * 128 + k0 + khalf;
                v2f a;
                a.x = hrow[kk];
                a.y = hrow[kk + 1];
                int kr = k0 + khalf;
#pragma unroll
                for (int nt = 0; nt < 4; ++nt) {
                    int col = nt * 16 + (lane & 15);
                    v2f b;
                    b.x = sW[kr * 64 + col];
                    b.y = sW[(kr + 1) * 64 + col];
                    acc[nt] = __builtin_amdgcn_wmma_f32_16x16x4_f32(
                        false, a, false, b, (short)0, acc[nt], false, false);
                }
            }
        }
        __syncthreads();
    }
    if (valid) {
        int mhi = (lane >> 4) << 3;       // C/D layout: VGPR r = rows r / r+8
#pragma unroll
        for (int nt = 0; nt < 4; ++nt) {
            int col = nt * 16 + (lane & 15);
#pragma unroll
            for (int r = 0; r < 8; ++r) {
                int m = m0 + r + mhi;
                xh2[(size_t)m * 64 + col] = acc[nt][r];
            }
        }
    }
}

// ---------------- K6: conv2 attention coefficients ----------------
__global__ void k_attn2(const float* __restrict__ xh2, const float* __restrict__ as2,
                        const float* __restrict__ ad2,
                        float* __restrict__ a2s, float* __restrict__ a2d) {
    int n    = (blockIdx.x * blockDim.x + threadIdx.x) >> 5;
    int lane = threadIdx.x & 31;
    if (n >= NNODES) return;
    size_t b = (size_t)n * 64;
    float v0 = xh2[b + lane], v1 = xh2[b + lane + 32];
    float ps = v0 * as2[lane] + v1 * as2[lane + 32];
    float pd = v0 * ad2[lane] + v1 * ad2[lane + 32];
    for (int off = 16; off; off >>= 1) {
        ps += __shfl_xor(ps, off, 32);
        pd += __shfl_xor(pd, off, 32);
    }
    if (lane == 0) { a2s[n] = ps; a2d[n] = pd; }
}

// ---------------- K7/K8: conv2 edge passes ----------------
__global__ void k_edge_max2(const int* __restrict__ ei, const float* __restrict__ ea,
                            const float* __restrict__ a2s, const float* __restrict__ a2d,
                            const float* __restrict__ ce2, unsigned* __restrict__ dmax2) {
    int e = blockIdx.x * blockDim.x + threadIdx.x;
    if (e >= NEDGES) return;
    int s = ei[e], d = ei[NEDGES + e];
    float lg = leaky(a2s[s] + a2d[d] + ea[e] * ce2[0]);
    atomicMax(&dmax2[d], f2ord(lg));
}

__global__ void k_edge_msg2(const int* __restrict__ ei, const float* __restrict__ ea,
                            const float* __restrict__ a2s, const float* __restrict__ a2d,
                            const float* __restrict__ ce2, const unsigned* __restrict__ dmax2,
                            const float* __restrict__ xh2,
                            float* __restrict__ dsum2, float* __restrict__ out2) {
    int e    = (blockIdx.x * blockDim.x + threadIdx.x) >> 5;
    int lane = threadIdx.x & 31;
    if (e >= NEDGES) return;
    int s = ei[e], d = ei[NEDGES + e];
    float lg = leaky(a2s[s] + a2d[d] + ea[e] * ce2[0]);
    float w  = __expf(lg - ord2f(dmax2[d]));
    if (lane == 0) atomicAdd(&dsum2[d], w);
    size_t sb = (size_t)s * 64, db = (size_t)d * 64;
    atomicAdd(&out2[db + lane],      xh2[sb + lane]      * w);
    atomicAdd(&out2[db + lane + 32], xh2[sb + lane + 32] * w);
}

// ---------------- K9: cluster mean pooling (hierarchical LDS reduction) ----------------
__global__ void k_pool(const float* __restrict__ out2, const float* __restrict__ dsum2,
                       const float* __restrict__ b2, const int* __restrict__ assign,
                       const float* __restrict__ x,
                       float* __restrict__ zsum, float* __restrict__ cfsum,
                       float* __restrict__ cnt) {
    __shared__ float lz[NCLUST * HID + 2 * NCLUST];   // 256 zsum + 4 cf + 4 counts
    int tid = threadIdx.x;
    for (int i = tid; i < NCLUST * HID + 2 * NCLUST; i += blockDim.x) lz[i] = 0.f;
    __syncthreads();
    int d   = tid & 63;
    int sub = tid >> 6;                               // 0..3 : 4 nodes per pass
    for (int it = 0; it < 16; ++it) {
        int n = blockIdx.x * 64 + it * 4 + sub;
        if (n < NNODES) {
            int cl = assign[n];
            float v = out2[(size_t)n * 64 + d] / dsum2[n] + b2[d];   // conv2 output
            atomicAdd(&lz[cl * HID + d], v);
            if (d == 0) {
                atomicAdd(&lz[NCLUST * HID + cl], x[(size_t)n * INDIM + 6]);
                atomicAdd(&lz[NCLUST * HID + NCLUST + cl], 1.f);
            }
        }
    }
    __syncthreads();
    for (int i = tid; i < NCLUST * HID; i += blockDim.x) atomicAdd(&zsum[i], lz[i]);
    if (tid < NCLUST)                    atomicAdd(&cfsum[tid], lz[NCLUST * HID + tid]);
    if (tid >= NCLUST && tid < 2*NCLUST) atomicAdd(&cnt[tid - NCLUST], lz[NCLUST * HID + tid]);
}

// ---------------- K10: actor/critic heads + softmax ----------------
__global__ void k_final(const float* __restrict__ zsum, const float* __restrict__ cfsum,
                        const float* __restrict__ cnt,
                        const float* __restrict__ A1, const float* __restrict__ ba1,
                        const float* __restrict__ A2, const float* __restrict__ ba2,
                        const float* __restrict__ C1, const float* __restrict__ bc1,
                        const float* __restrict__ C2, const float* __restrict__ bc2,
                        float* __restrict__ out) {
    __shared__ float sZ[256], sCF[4], sPA[256], sLG[4], sPC[64];
    int t = threadIdx.x;
    if (t < 256) {
        float c = cnt[t >> 6];
        float v = c > 0.f ? zsum[t] / fmaxf(c, 1.f) : 0.f;
        sZ[t] = v;
        out[5 + t] = v;                               // z_flat
    }
    if (t < 4) {
        float c = cnt[t];
        sCF[t] = c > 0.f ? cfsum[t] / fmaxf(c, 1.f) : 0.f;
    }
    __syncthreads();
    if (t < 64) {
#pragma unroll
        for (int c = 0; c < 4; ++c) {                 // actor hidden
            float hv = ba1[t];
            for (int i = 0; i < 64; ++i) hv += sZ[c * 64 + i] * A1[i * 64 + t];
            hv += sCF[c] * A1[64 * 64 + t];
            hv = fmaxf(hv, 0.f);
            sPA[c * 64 + t] = hv * A2[t];
        }
        float hc = bc1[t];                            // critic hidden
        for (int i = 0; i < 256; ++i) hc += sZ[i] * C1[i * 64 + t];
        hc = fmaxf(hc, 0.f);
        sPC[t] = hc * C2[t];
    }
    __syncthreads();
    if (t < 4) {
        float lg = ba2[0];
        for (int i = 0; i < 64; ++i) lg += sPA[t * 64 + i];
        sLG[t] = lg;
    }
    __syncthreads();
    if (t == 0) {
        float v = bc2[0];
        for (int i = 0; i < 64; ++i) v += sPC[i];
        out[4] = v;                                   // value
        float m = fmaxf(fmaxf(sLG[0], sLG[1]), fmaxf(sLG[2], sLG[3]));
        float ex[4], ssum = 0.f;
#pragma unroll
        for (int c = 0; c < 4; ++c) { ex[c] = __expf(sLG[c] - m); ssum += ex[c]; }
#pragma unroll
        for (int c = 0; c < 4; ++c) out[c] = ex[c] / ssum;   // probs
    }
}

// ---------------- launch ----------------
extern "C" void kernel_launch(void* const* d_in, const int* in_sizes, int n_in,
                              void* d_out, int out_size, void* d_ws, size_t ws_size,
                              hipStream_t stream) {
    (void)in_sizes; (void)n_in; (void)out_size; (void)ws_size;
    const float* x    = (const float*)d_in[0];
    const int*   ei   = (const int*)  d_in[1];
    const float* ea   = (const float*)d_in[2];
    const int*   asg  = (const int*)  d_in[3];
    const float* W1   = (const float*)d_in[4];
    const float* as1  = (const float*)d_in[5];
    const float* ad1  = (const float*)d_in[6];
    const float* We1  = (const float*)d_in[7];
    const float* ae1  = (const float*)d_in[8];
    const float* b1   = (const float*)d_in[9];
    const float* W2   = (const float*)d_in[10];
    const float* as2  = (const float*)d_in[11];
    const float* ad2  = (const float*)d_in[12];
    const float* We2  = (const float*)d_in[13];
    const float* ae2  = (const float*)d_in[14];
    const float* b2   = (const float*)d_in[15];
    const float* A1   = (const float*)d_in[16];
    const float* ba1  = (const float*)d_in[17];
    const float* A2   = (const float*)d_in[18];
    const float* ba2  = (const float*)d_in[19];
    const float* C1   = (const float*)d_in[20];
    const float* bc1  = (const float*)d_in[21];
    const float* C2   = (const float*)d_in[22];
    const float* bc2  = (const float*)d_in[23];
    float* out = (float*)d_out;

    float* ws = (float*)d_ws;
    size_t o = 0;
    float*    xh1   = ws + o; o += (size_t)NNODES * 256;
    float*    acc1  = ws + o; o += (size_t)NNODES * 256;  // msg accumulator -> h1 in place
    float*    asrc1 = ws + o; o += (size_t)NNODES * 4;
    float*    adst1 = ws + o; o += (size_t)NNODES * 4;
    float*    dsum1 = ws + o; o += (size_t)NNODES * 4;
    unsigned* dmax1 = (unsigned*)(ws + o); o += (size_t)NNODES * 4;
    float*    xh2   = ws + o; o += (size_t)NNODES * 64;
    float*    out2  = ws + o; o += (size_t)NNODES * 64;
    float*    a2s   = ws + o; o += NNODES;
    float*    a2d   = ws + o; o += NNODES;
    float*    dsum2 = ws + o; o += NNODES;
    unsigned* dmax2 = (unsigned*)(ws + o); o += NNODES;
    float*    zsum  = ws + o; o += 256;
    float*    cfsum = ws + o; o += 4;
    float*    cnt   = ws + o; o += 4;
    float*    ce1   = ws + o; o += 4;
    float*    ce2   = ws + o; o += 1;

    k_setup    <<<1, 32, 0, stream>>>(We1, ae1, We2, ae2, ce1, ce2);
    k_node1    <<<(NNODES + 7) / 8, 256, 0, stream>>>(x, W1, as1, ad1, xh1, acc1,
                                                      asrc1, adst1, dsum1, dmax1);
    k_edge_max1<<<(NEDGES * 4 + 255) / 256, 256, 0, stream>>>(ei, ea, asrc1, adst1, ce1, dmax1);
    k_edge_msg1<<<(NEDGES + 7) / 8, 256, 0, stream>>>(ei, ea, asrc1, adst1, ce1, dmax1,
                                                      xh1, dsum1, acc1);
    k_finish1  <<<(NNODES + 7) / 8, 256, 0, stream>>>(dsum1, b1, acc1, out2, dsum2, dmax2,
                                                      zsum, cfsum, cnt);
    k_gemm2    <<<(MTILES + 7) / 8, 256, 0, stream>>>(acc1, W2, xh2);
    k_attn2    <<<(NNODES + 7) / 8, 256, 0, stream>>>(xh2, as2, ad2, a2s, a2d);
    k_edge_max2<<<(NEDGES + 255) / 256, 256, 0, stream>>>(ei, ea, a2s, a2d, ce2, dmax2);
    k_edge_msg2<<<(NEDGES + 7) / 8, 256, 0, stream>>>(ei, ea, a2s, a2d, ce2, dmax2,
                                                      xh2, dsum2, out2);
    k_pool     <<<(NNODES + 63) / 64, 256, 0, stream>>>(out2, dsum2, b2, asg, x,
                                                        zsum, cfsum, cnt);
    k_final    <<<1, 256, 0, stream>>>(zsum, cfsum, cnt, A1, ba1, A2, ba2,
                                       C1, bc1, C2, bc2, out);
}